// HGT_13056700580219
// MI455X (gfx1250) — compile-verified
//
#include <hip/hip_runtime.h>
#include <hip/hip_bf16.h>
#include <math.h>

// ---------------------------------------------------------------------------
// HGT forward for MI455X (gfx1250). Dense matmuls use V_WMMA_F32_16X16X4_F32
// (wave32). Each GEMM wave computes a 16x64 output tile (4 accumulators) so
// the A fragment is reused 4x per K-step. Edge softmax/agg uses f32 atomics.
// ---------------------------------------------------------------------------

#define NQ 10000
#define NC 40000
#define CDIM 128
#define HH 8
#define DD 16
#define EQC 160000
#define ECQ 160000
#define ECC 320000

typedef __attribute__((ext_vector_type(2))) float v2f;
typedef __attribute__((ext_vector_type(8))) float v8f;

// ---------------------------------------------------------------------------
// WMMA f32 GEMM:  Out[M,N] = epi( A[M,K] @ W[K,N] + bias )
// One wave -> one 16x64 tile (4x v8f accumulators), blockDim=(32,4).
// M mult of 16, N mult of 64, K mult of 4.
// act: 0 none, 1 relu. If xprev != null: out = g*val + (1-g)*xprev,
// g = sigmoid(*skip).
// ---------------------------------------------------------------------------
__global__ __launch_bounds__(128)
void wmma_gemm(const float* __restrict__ A, const float* __restrict__ W,
               const float* __restrict__ bias, float* __restrict__ Out,
               int M, int N, int K, int act,
               const float* __restrict__ xprev, const float* __restrict__ skip)
{
    const int lane   = threadIdx.x;          // 0..31
    const int tilesN = N >> 6;               // 64-wide wave tiles
    const int tile   = blockIdx.x * 4 + threadIdx.y;
    const int numTiles = (M >> 4) * tilesN;
    if (tile >= numTiles) return;            // wave-uniform (EXEC stays all-1)
    const int tm = (tile / tilesN) << 4;
    const int tn = (tile % tilesN) << 6;

    const int half = lane >> 4;              // 0: K rows 0/1, 1: K rows 2/3
    const int l15  = lane & 15;
    const int arow = tm + l15;               // A row this lane loads
    const int bc0  = tn + l15;               // first of 4 output columns

    v8f acc[4] = {{}, {}, {}, {}};
    #pragma unroll 2
    for (int kk = 0; kk < K; kk += 4) {
        const int kb = kk + (half << 1);
        v2f a; a.x = A[arow * K + kb]; a.y = A[arow * K + kb + 1];
        const float* __restrict__ w0 = W + (size_t)kb * N + bc0;   // row kb
        #pragma unroll
        for (int j = 0; j < 4; ++j) {
            v2f b; b.x = w0[j * 16]; b.y = w0[N + j * 16];         // rows kb,kb+1
            acc[j] = __builtin_amdgcn_wmma_f32_16x16x4_f32(
                         false, a, false, b, (short)0, acc[j], false, false);
        }
    }

    float g = 1.0f, og = 0.0f;
    if (xprev) { float sv = skip[0]; g = 1.0f / (1.0f + __expf(-sv)); og = 1.0f - g; }

    #pragma unroll
    for (int j = 0; j < 4; ++j) {
        const int col = bc0 + j * 16;
        const float bcv = bias ? bias[col] : 0.0f;
        #pragma unroll
        for (int r = 0; r < 8; ++r) {
            const int row = tm + r + (half << 3);
            float v = acc[j][r] + bcv;
            if (act == 1) v = fmaxf(v, 0.0f);
            if (xprev)   v = g * v + og * xprev[(size_t)row * N + col];
            Out[(size_t)row * N + col] = v;
        }
    }
}

// ---------------------------------------------------------------------------
// Relation transform: Out[n, h*16+e] = sum_d X[n, h*16+d] * AM[h][d][e]
// One wave -> (16-node tile, one head); K=16 -> four 16x16x4 WMMA steps.
// ---------------------------------------------------------------------------
__global__ __launch_bounds__(128)
void wmma_rel(const float* __restrict__ X, const float* __restrict__ AM,
              float* __restrict__ Out, int Nnodes)
{
    const int lane = threadIdx.x;
    const int tile = blockIdx.x * 4 + threadIdx.y;
    const int numTiles = (Nnodes >> 4) * HH;
    if (tile >= numTiles) return;
    const int tm = (tile >> 3) << 4;        // node tile base
    const int h  = tile & 7;
    const int half = lane >> 4, l15 = lane & 15;
    const int arow = tm + l15;
    const float* __restrict__ Ah = AM + h * 256;

    v8f acc = {};
    #pragma unroll
    for (int kk = 0; kk < 16; kk += 4) {
        const int kb = kk + (half << 1);
        v2f a; a.x = X[arow * CDIM + h * DD + kb];
               a.y = X[arow * CDIM + h * DD + kb + 1];
        v2f b; b.x = Ah[kb * DD + l15];
               b.y = Ah[(kb + 1) * DD + l15];
        acc = __builtin_amdgcn_wmma_f32_16x16x4_f32(
                  false, a, false, b, (short)0, acc, false, false);
    }
    #pragma unroll
    for (int r = 0; r < 8; ++r)
        Out[(tm + r + (half << 3)) * CDIM + h * DD + l15] = acc[r];
}

// -------------------- BatchNorm stats (coalesced, LDS+atomics) -------------
__global__ void colstats_part(const float* __restrict__ H, int total,
                              float* __restrict__ sums, float* __restrict__ sqs)
{
    __shared__ float ls[CDIM], lq[CDIM];
    if (threadIdx.x < CDIM) { ls[threadIdx.x] = 0.f; lq[threadIdx.x] = 0.f; }
    __syncthreads();
    for (int i = blockIdx.x * blockDim.x + threadIdx.x; i < total;
         i += gridDim.x * blockDim.x) {
        const float v = H[i];
        atomicAdd(&ls[i & (CDIM - 1)], v);       // ds_add_f32, bank-conflict free in-wave
        atomicAdd(&lq[i & (CDIM - 1)], v * v);
    }
    __syncthreads();
    if (threadIdx.x < CDIM) {
        atomicAdd(&sums[threadIdx.x], ls[threadIdx.x]);
        atomicAdd(&sqs[threadIdx.x],  lq[threadIdx.x]);
    }
}

__global__ void bn_apply(float* __restrict__ X, const float* __restrict__ sums,
                         const float* __restrict__ sqs, const float* __restrict__ g,
                         const float* __restrict__ be, int Nrows, int total)
{
    int i = blockIdx.x * blockDim.x + threadIdx.x;
    if (i >= total) return;
    const int c = i & (CDIM - 1);
    const float mu  = sums[c] / (float)Nrows;
    const float var = sqs[c] / (float)Nrows - mu * mu;
    X[i] = (X[i] - mu) * rsqrtf(var + 1e-5f) * g[c] + be[c];
}

// ------------------------------ utilities ----------------------------------
__global__ void fillf(float* __restrict__ p, float v, int n)
{
    int i = blockIdx.x * blockDim.x + threadIdx.x;
    if (i < n) p[i] = v;
}

__device__ __forceinline__ void atomicMaxFloat(float* addr, float val)
{
    if (val >= 0.0f) atomicMax((int*)addr, __float_as_int(val));
    else             atomicMin((unsigned int*)addr, __float_as_uint(val));
}

// -------------------------- edge pass A: logits + max ----------------------
__global__ void edge_logits(const int* __restrict__ src, const int* __restrict__ dst,
                            const float* __restrict__ Q, const float* __restrict__ Krel,
                            const float* __restrict__ P, float* __restrict__ logits,
                            float* __restrict__ mbuf, int E)
{
    int tid = blockIdx.x * blockDim.x + threadIdx.x;
    if (tid >= E * HH) return;
    const int e = tid >> 3, h = tid & 7;
    const int sN = src[e], dN = dst[e];
    const float* __restrict__ qv = Q    + (size_t)dN * CDIM + h * DD;
    const float* __restrict__ kv = Krel + (size_t)sN * CDIM + h * DD;
    float acc = 0.f;
    #pragma unroll
    for (int d = 0; d < DD; ++d) acc += qv[d] * kv[d];
    const float l = acc * P[h] * 0.25f;     // / sqrt(D=16)
    logits[tid] = l;
    atomicMaxFloat(mbuf + (size_t)dN * HH + h, l);
}

// ------------------- edge pass B: exp, sum(e), sum(e*v) --------------------
__global__ void edge_accum(const int* __restrict__ src, const int* __restrict__ dst,
                           const float* __restrict__ Vrel, const float* __restrict__ logits,
                           const float* __restrict__ mbuf, float* __restrict__ sbuf,
                           float* __restrict__ agg, int E)
{
    int tid = blockIdx.x * blockDim.x + threadIdx.x;
    if (tid >= E * HH) return;
    const int e = tid >> 3, h = tid & 7;
    const int sN = src[e], dN = dst[e];
    const float ex = __expf(logits[tid] - mbuf[(size_t)dN * HH + h]);
    atomicAdd(sbuf + (size_t)dN * HH + h, ex);
    const float* __restrict__ vv = Vrel + (size_t)sN * CDIM + h * DD;
    float* __restrict__ av = agg + (size_t)dN * CDIM + h * DD;
    #pragma unroll
    for (int d = 0; d < DD; ++d) atomicAdd(av + d, ex * vv[d]);
}

// ------------------ softmax divide + exact GELU (in place) -----------------
__global__ void gelu_div(float* __restrict__ agg, const float* __restrict__ sbuf, int total)
{
    int i = blockIdx.x * blockDim.x + threadIdx.x;
    if (i >= total) return;
    const int n = i >> 7, h = (i & (CDIM - 1)) >> 4;
    const float sv = sbuf[(size_t)n * HH + h];
    const float x = (sv > 0.f) ? agg[i] / sv : 0.f;
    agg[i] = 0.5f * x * (1.0f + erff(x * 0.70710678118654752f));
}

// ===========================================================================
extern "C" void kernel_launch(void* const* d_in, const int* in_sizes, int n_in,
                              void* d_out, int out_size, void* d_ws, size_t ws_size,
                              hipStream_t stream)
{
    (void)in_sizes; (void)n_in; (void)out_size; (void)ws_size;

    // ---- top-level inputs (setup_inputs dict order) ----
    const float* Xq     = (const float*)d_in[0];   // [NQ,768]
    const float* Xc     = (const float*)d_in[1];   // [NC,128]
    const int*   src_qc = (const int*)d_in[2];
    const int*   dst_qc = (const int*)d_in[3];
    const int*   src_cq = (const int*)d_in[4];
    const int*   dst_cq = (const int*)d_in[5];
    const int*   src_cc = (const int*)d_in[6];
    const int*   dst_cc = (const int*)d_in[7];

    // ---- params, JAX pytree flatten order (dict keys sorted) ----
    // params: bn < layers < lin ; node types: concept < question ;
    // kqv keys: Wk<Wq<Wv<bk<bq<bv ; rel keys: cc<cq<qc ; rel inner: a<m<p
    const float* bn_c_g = (const float*)d_in[8];
    const float* bn_c_b = (const float*)d_in[9];
    const float* bn_q_g = (const float*)d_in[10];
    const float* bn_q_b = (const float*)d_in[11];

    struct LayerP {
        const float *WkC,*WqC,*WvC,*bkC,*bqC,*bvC;
        const float *WkQ,*WqQ,*WvQ,*bkQ,*bqQ,*bvQ;
        const float *WoC,*boC,*WoQ,*boQ;
        const float *a_cc,*m_cc,*p_cc, *a_cq,*m_cq,*p_cq, *a_qc,*m_qc,*p_qc;
        const float *skipC,*skipQ;
    } LP[2];
    for (int l = 0; l < 2; ++l) {
        const int b = 12 + l * 27;
        LP[l].WkC=(const float*)d_in[b+0];  LP[l].WqC=(const float*)d_in[b+1];
        LP[l].WvC=(const float*)d_in[b+2];  LP[l].bkC=(const float*)d_in[b+3];
        LP[l].bqC=(const float*)d_in[b+4];  LP[l].bvC=(const float*)d_in[b+5];
        LP[l].WkQ=(const float*)d_in[b+6];  LP[l].WqQ=(const float*)d_in[b+7];
        LP[l].WvQ=(const float*)d_in[b+8];  LP[l].bkQ=(const float*)d_in[b+9];
        LP[l].bqQ=(const float*)d_in[b+10]; LP[l].bvQ=(const float*)d_in[b+11];
        LP[l].WoC=(const float*)d_in[b+12]; LP[l].boC=(const float*)d_in[b+13];
        LP[l].WoQ=(const float*)d_in[b+14]; LP[l].boQ=(const float*)d_in[b+15];
        LP[l].a_cc=(const float*)d_in[b+16]; LP[l].m_cc=(const float*)d_in[b+17];
        LP[l].p_cc=(const float*)d_in[b+18];
        LP[l].a_cq=(const float*)d_in[b+19]; LP[l].m_cq=(const float*)d_in[b+20];
        LP[l].p_cq=(const float*)d_in[b+21];
        LP[l].a_qc=(const float*)d_in[b+22]; LP[l].m_qc=(const float*)d_in[b+23];
        LP[l].p_qc=(const float*)d_in[b+24];
        LP[l].skipC=(const float*)d_in[b+25]; LP[l].skipQ=(const float*)d_in[b+26];
    }
    const float* lin_c_W = (const float*)d_in[66];
    const float* lin_c_b = (const float*)d_in[67];
    const float* lin_q_W = (const float*)d_in[68];
    const float* lin_q_b = (const float*)d_in[69];

    // ---- workspace layout (floats) ----
    float* ws = (float*)d_ws;
    size_t o = 0;
    auto alloc = [&](size_t n){ float* p = ws + o; o += n; return p; };
    float* xq   = alloc((size_t)NQ * CDIM);
    float* xc   = alloc((size_t)NC * CDIM);
    float* xq2  = alloc((size_t)NQ * CDIM);
    float* xc2  = alloc((size_t)NC * CDIM);
    float* qq   = alloc((size_t)NQ * CDIM);
    float* kq   = alloc((size_t)NQ * CDIM);
    float* vq   = alloc((size_t)NQ * CDIM);
    float* qc_  = alloc((size_t)NC * CDIM);
    float* kc   = alloc((size_t)NC * CDIM);
    float* vc   = alloc((size_t)NC * CDIM);
    float* krel = alloc((size_t)NC * CDIM);
    float* vrelA= alloc((size_t)NC * CDIM);
    float* vrelB= alloc((size_t)NC * CDIM);
    float* lgt  = alloc((size_t)(EQC + ECC) * HH);   // worst-case dest=concept
    float* mb   = alloc((size_t)NC * HH);
    float* sb   = alloc((size_t)NC * HH);
    float* agg  = alloc((size_t)NC * CDIM);
    float* sums = alloc(CDIM);
    float* sqs  = alloc(CDIM);

    const dim3 blk128(32, 4);
    auto gemm = [&](const float* A, const float* W, const float* bias, float* O,
                    int M, int N, int K, int act,
                    const float* xprev, const float* skip){
        int tiles = (M >> 4) * (N >> 6);          // 16x64 wave tiles
        wmma_gemm<<<dim3((tiles + 3) / 4), blk128, 0, stream>>>(
            A, W, bias, O, M, N, K, act, xprev, skip);
    };
    auto rel = [&](const float* X, const float* AM, float* O, int N){
        int tiles = (N >> 4) * HH;
        wmma_rel<<<dim3((tiles + 3) / 4), blk128, 0, stream>>>(X, AM, O, N);
    };
    auto fill = [&](float* p, float v, int n){
        fillf<<<dim3((n + 255) / 256), dim3(256), 0, stream>>>(p, v, n);
    };
    auto batchnorm = [&](float* X, const float* g, const float* be, int Nrows){
        const int total = Nrows * CDIM;
        fill(sums, 0.0f, CDIM);
        fill(sqs,  0.0f, CDIM);
        colstats_part<<<dim3(512), dim3(256), 0, stream>>>(X, total, sums, sqs);
        bn_apply<<<dim3((total + 255)/256), dim3(256), 0, stream>>>(
            X, sums, sqs, g, be, Nrows, total);
    };

    // ================= stage 0: lin + relu + batchnorm =================
    gemm(Xq, lin_q_W, lin_q_b, xq, NQ, CDIM, 768, /*relu*/1, nullptr, nullptr);
    batchnorm(xq, bn_q_g, bn_q_b, NQ);

    gemm(Xc, lin_c_W, lin_c_b, xc, NC, CDIM, CDIM, /*relu*/1, nullptr, nullptr);
    batchnorm(xc, bn_c_g, bn_c_b, NC);

    // ================= HGT layers =================
    float* cur_q = xq; float* cur_c = xc;
    for (int l = 0; l < 2; ++l) {
        const LayerP& P = LP[l];
        float* outq = (l == 1) ? (float*)d_out : xq2;
        float* outc = (l == 1) ? ((float*)d_out + (size_t)NQ * CDIM) : xc2;

        // K/Q/V projections
        gemm(cur_q, P.WqQ, P.bqQ, qq,  NQ, CDIM, CDIM, 0, nullptr, nullptr);
        gemm(cur_q, P.WkQ, P.bkQ, kq,  NQ, CDIM, CDIM, 0, nullptr, nullptr);
        gemm(cur_q, P.WvQ, P.bvQ, vq,  NQ, CDIM, CDIM, 0, nullptr, nullptr);
        gemm(cur_c, P.WqC, P.bqC, qc_, NC, CDIM, CDIM, 0, nullptr, nullptr);
        gemm(cur_c, P.WkC, P.bkC, kc,  NC, CDIM, CDIM, 0, nullptr, nullptr);
        gemm(cur_c, P.WvC, P.bvC, vc,  NC, CDIM, CDIM, 0, nullptr, nullptr);

        // ----- dest = question (relation cq, src = concept) -----
        fill(mb, -INFINITY, NQ * HH);
        fill(sb, 0.0f,      NQ * HH);
        fill(agg, 0.0f,     NQ * CDIM);
        rel(kc, P.a_cq, krel,  NC);
        rel(vc, P.m_cq, vrelA, NC);
        edge_logits<<<dim3((ECQ*HH + 255)/256), dim3(256), 0, stream>>>(
            src_cq, dst_cq, qq, krel, P.p_cq, lgt, mb, ECQ);
        edge_accum<<<dim3((ECQ*HH + 255)/256), dim3(256), 0, stream>>>(
            src_cq, dst_cq, vrelA, lgt, mb, sb, agg, ECQ);
        gelu_div<<<dim3((NQ*CDIM + 255)/256), dim3(256), 0, stream>>>(agg, sb, NQ * CDIM);
        gemm(agg, P.WoQ, P.boQ, outq, NQ, CDIM, CDIM, 0, cur_q, P.skipQ);

        // ----- dest = concept (relations qc src=question, cc src=concept) -----
        fill(mb, -INFINITY, NC * HH);
        fill(sb, 0.0f,      NC * HH);
        fill(agg, 0.0f,     NC * CDIM);
        rel(kq, P.a_qc, krel,  NQ);
        rel(vq, P.m_qc, vrelA, NQ);
        edge_logits<<<dim3((EQC*HH + 255)/256), dim3(256), 0, stream>>>(
            src_qc, dst_qc, qc_, krel, P.p_qc, lgt, mb, EQC);
        rel(kc, P.a_cc, krel,  NC);
        rel(vc, P.m_cc, vrelB, NC);
        edge_logits<<<dim3((ECC*HH + 255)/256), dim3(256), 0, stream>>>(
            src_cc, dst_cc, qc_, krel, P.p_cc, lgt + (size_t)EQC * HH, mb, ECC);
        edge_accum<<<dim3((EQC*HH + 255)/256), dim3(256), 0, stream>>>(
            src_qc, dst_qc, vrelA, lgt, mb, sb, agg, EQC);
        edge_accum<<<dim3((ECC*HH + 255)/256), dim3(256), 0, stream>>>(
            src_cc, dst_cc, vrelB, lgt + (size_t)EQC * HH, mb, sb, agg, ECC);
        gelu_div<<<dim3((NC*CDIM + 255)/256), dim3(256), 0, stream>>>(agg, sb, NC * CDIM);
        gemm(agg, P.WoC, P.boC, outc, NC, CDIM, CDIM, 0, cur_c, P.skipC);

        cur_q = outq; cur_c = outc;
    }
}